// HR2HK_20572893348009
// MI455X (gfx1250) — compile-verified
//
#include <hip/hip_runtime.h>
#include <hip/hip_bf16.h>
#include <math.h>

#define A_     96
#define E_     2048
#define K_     16
#define N_     9
#define F_     1728
#define AN_    864
#define FEAT_  648
#define PAIRS_ 9216

typedef __attribute__((ext_vector_type(2))) float v2f;
typedef __attribute__((ext_vector_type(8))) float v8f;

// ---- feature-index math derived from _build_blocks ----
// DIMS = (1,3,5); block (ii,jj) starts at st = 8*(9*pref[ii] + di*pref[jj]),
// within block: real idx = st + ((s1*2+s2)*di + p)*dj + q, imag = real + 4*di*dj.
__device__ __forceinline__ void blk_of(int n, int& p, int& d, int& pref) {
    if (n < 1)      { p = n;     d = 1; pref = 0; }
    else if (n < 4) { p = n - 1; d = 3; pref = 1; }
    else            { p = n - 4; d = 5; pref = 4; }
}

__device__ __forceinline__ void feat_idx(int si, int ni, int sj, int nj,
                                         int& ridx, int& iidx) {
    int p, di, pa, q, dj, pb;
    blk_of(ni, p, di, pa);
    blk_of(nj, q, dj, pb);
    int st = 8 * (9 * pa + di * pb);
    ridx = st + ((si * 2 + sj) * di + p) * dj + q;
    iidx = ridx + 4 * di * dj;
}

// ---- Kernel 1: phase[e,k] = exp(-i * 2pi * shift[e] . kpt[k]) via WMMA ----
// One wave32 per 16 edges: ang(16x16) = A(16x4) x B(4x16) with
// A[m,d] = shift[e0+m, d] (d<3, pad 0), B[d,n] = 2pi*kpt[n,d].
__global__ void phase_kernel(const float* __restrict__ shift,
                             const float* __restrict__ kpt,
                             float2* __restrict__ pw) {
    const int lane = threadIdx.x;         // 0..31
    const int e0   = blockIdx.x * 16;
    const int m    = lane & 15;
    const int hi   = lane >> 4;           // 0: K=0,1  1: K=2,3
    const float TP = 6.283185307179586f;
#if __has_builtin(__builtin_amdgcn_wmma_f32_16x16x4_f32)
    const int e = e0 + m;
    v2f a, b;
    // 32-bit A 16x4 layout: lanes 0-15 -> K=0 (v0), K=1 (v1); lanes 16-31 -> K=2,3
    a.x = hi ? shift[e * 3 + 2] : shift[e * 3 + 0];
    a.y = hi ? 0.0f             : shift[e * 3 + 1];
    // B 4x16: row striped across lanes; lanes 0-15 rows 0/1, lanes 16-31 rows 2/3
    b.x = TP * (hi ? kpt[m * 3 + 2] : kpt[m * 3 + 0]);
    b.y = hi ? 0.0f : TP * kpt[m * 3 + 1];
    v8f c = {0.f, 0.f, 0.f, 0.f, 0.f, 0.f, 0.f, 0.f};
    c = __builtin_amdgcn_wmma_f32_16x16x4_f32(false, a, false, b,
                                              (short)0, c, false, false);
    // D 16x16 f32: VGPR r -> M = r (lanes 0-15) or 8+r (lanes 16-31), N = lane&15
#pragma unroll
    for (int r = 0; r < 8; ++r) {
        int row = r + hi * 8;             // edge index within tile
        float sn, cs;
        sincosf(c[r], &sn, &cs);
        pw[(size_t)(e0 + row) * K_ + m] = make_float2(cs, -sn);
    }
#else
    for (int i = lane; i < 16 * K_; i += 32) {
        int mm = i / K_, kk = i % K_;
        int e = e0 + mm;
        float ang = TP * (shift[e * 3 + 0] * kpt[kk * 3 + 0] +
                          shift[e * 3 + 1] * kpt[kk * 3 + 1] +
                          shift[e * 3 + 2] * kpt[kk * 3 + 2]);
        float sn, cs;
        sincosf(ang, &sn, &cs);
        pw[(size_t)e * K_ + kk] = make_float2(cs, -sn);
    }
#endif
}

// ---- CSR build over atom pairs (deterministic) ----
__global__ void zero_counts_kernel(int* __restrict__ counts) {
    int i = blockIdx.x * blockDim.x + threadIdx.x;
    if (i < PAIRS_) counts[i] = 0;
}

__global__ void count_kernel(const int* __restrict__ eidx, int* __restrict__ counts) {
    int e = blockIdx.x * blockDim.x + threadIdx.x;
    if (e < E_) {
        int key = eidx[e] * A_ + eidx[E_ + e];
        atomicAdd(&counts[key], 1);   // integer: order-independent -> deterministic
    }
}

__global__ void scan_kernel(const int* __restrict__ counts, int* __restrict__ offs) {
    __shared__ int tot[1024];
    __shared__ int pref[1025];
    const int tid  = threadIdx.x;
    const int base = tid * 9;                 // 9216 / 1024 = 9
    int s = 0;
#pragma unroll
    for (int i = 0; i < 9; ++i) s += counts[base + i];
    tot[tid] = s;
    __syncthreads();
    if (tid == 0) {
        int r = 0;
        for (int i = 0; i < 1024; ++i) { pref[i] = r; r += tot[i]; }
        pref[1024] = r;
    }
    __syncthreads();
    int run = pref[tid];
#pragma unroll
    for (int i = 0; i < 9; ++i) { offs[base + i] = run; run += counts[base + i]; }
    if (tid == 1023) offs[PAIRS_] = pref[1024];
}

// rank of edge e = #earlier edges with same pair -> list sorted by e (deterministic)
__global__ void fill_kernel(const int* __restrict__ eidx,
                            const int* __restrict__ offs,
                            int* __restrict__ lists) {
    __shared__ int key[E_];
    for (int i = threadIdx.x; i < E_; i += blockDim.x)
        key[i] = eidx[i] * A_ + eidx[E_ + i];
    __syncthreads();
    int e = blockIdx.x * 128 + threadIdx.x;
    if (threadIdx.x < 128 && e < E_) {
        int mk = key[e];
        int rank = 0;
        for (int e2 = 0; e2 < e; ++e2) rank += (key[e2] == mk);
        lists[offs[mk] + rank] = e;
    }
}

// ---- Main kernel: one block owns one (pair, k) 18x18 complex output block ----
__global__ void __launch_bounds__(128)
main_kernel(const float* __restrict__ hop, const float* __restrict__ ons,
            const float2* __restrict__ pw, const int* __restrict__ offs,
            const int* __restrict__ lists, float* __restrict__ out) {
    const int pair = blockIdx.x;
    const int k    = blockIdx.y;
    const int a1   = pair / A_;
    const int a2   = pair - a1 * A_;
    const int tid  = threadIdx.x;

    int ridxA[3], iidxA[3], ridxB[3], iidxB[3];
    size_t ooff[3];
    bool   valid[3];
    float  accr[3], acci[3];

#pragma unroll
    for (int s = 0; s < 3; ++s) {
        int el   = tid + s * 128;
        valid[s] = (el < 324);
        accr[s]  = 0.0f;
        acci[s]  = 0.0f;
        int e    = valid[s] ? el : 0;
        // el = ((s1*9 + n1)*2 + s2)*9 + n2
        int n2 = e % 9;  int t = e / 9;
        int s2 = t & 1;  t >>= 1;
        int n1 = t % 9;  int s1 = t / 9;
        feat_idx(s1, n1, s2, n2, ridxA[s], iidxA[s]);   // direct h[s1,n1,s2,n2]
        feat_idx(s2, n2, s1, n1, ridxB[s], iidxB[s]);   // transposed h[s2,n2,s1,n1]
        int r = s1 * AN_ + a1 * N_ + n1;
        int c = s2 * AN_ + a2 * N_ + n2;
        ooff[s] = (((size_t)k * F_ + r) * F_ + c) * 2;
    }

    const int d0 = offs[pair],          d1 = offs[pair + 1];
    const int pr = a2 * A_ + a1;
    const int r0 = offs[pr],            r1 = offs[pr + 1];

    // direct edges: (src=a1, dst=a2): += 0.5*(hr + i*hi) * (px + i*py)
    for (int q = d0; q < d1; ++q) {
        int e = lists[q];
        float2 ph = pw[(size_t)e * K_ + k];
        const float* f = hop + (size_t)e * FEAT_;
#pragma unroll
        for (int s = 0; s < 3; ++s) {
            float hr = 0.5f * f[ridxA[s]];
            float hi = 0.5f * f[iidxA[s]];
            accr[s] += hr * ph.x - hi * ph.y;
            acci[s] += hr * ph.y + hi * ph.x;
        }
    }
    // reverse edges: (src=a2, dst=a1): += conj( h[s2,n2,s1,n1] * phase )
    for (int q = r0; q < r1; ++q) {
        int e = lists[q];
        float2 ph = pw[(size_t)e * K_ + k];
        const float* f = hop + (size_t)e * FEAT_;
#pragma unroll
        for (int s = 0; s < 3; ++s) {
            float hr = 0.5f * f[ridxB[s]];
            float hi = 0.5f * f[iidxB[s]];
            accr[s] += hr * ph.x - hi * ph.y;
            acci[s] -= hr * ph.y + hi * ph.x;
        }
    }
    // onsite diagonal: += ons(s1,n1,s2,n2) + conj(ons(s2,n2,s1,n1))
    if (a1 == a2) {
        const float* f = ons + (size_t)a1 * FEAT_;
#pragma unroll
        for (int s = 0; s < 3; ++s) {
            accr[s] += 0.5f * (f[ridxA[s]] + f[ridxB[s]]);
            acci[s] += 0.5f * (f[iidxA[s]] - f[iidxB[s]]);
        }
    }
#pragma unroll
    for (int s = 0; s < 3; ++s) {
        if (valid[s]) {
            v2f v;
            v.x = accr[s];
            v.y = acci[s];
            __builtin_nontemporal_store(v, (v2f*)(out + ooff[s]));
        }
    }
}

// ---- Fallback (no scratch): full edge scan, inline sincos ----
__global__ void __launch_bounds__(128)
fallback_kernel(const float* __restrict__ hop, const float* __restrict__ ons,
                const float* __restrict__ kpt, const int* __restrict__ eidx,
                const float* __restrict__ shift, float* __restrict__ out) {
    const int pair = blockIdx.x;
    const int k    = blockIdx.y;
    const int a1   = pair / A_;
    const int a2   = pair - a1 * A_;
    const int tid  = threadIdx.x;
    const float TP = 6.283185307179586f;
    const float kx = TP * kpt[k * 3 + 0];
    const float ky = TP * kpt[k * 3 + 1];
    const float kz = TP * kpt[k * 3 + 2];

    int ridxA[3], iidxA[3], ridxB[3], iidxB[3];
    size_t ooff[3];
    bool   valid[3];
    float  accr[3], acci[3];
#pragma unroll
    for (int s = 0; s < 3; ++s) {
        int el   = tid + s * 128;
        valid[s] = (el < 324);
        accr[s]  = 0.0f;  acci[s] = 0.0f;
        int e    = valid[s] ? el : 0;
        int n2 = e % 9;  int t = e / 9;
        int s2 = t & 1;  t >>= 1;
        int n1 = t % 9;  int s1 = t / 9;
        feat_idx(s1, n1, s2, n2, ridxA[s], iidxA[s]);
        feat_idx(s2, n2, s1, n1, ridxB[s], iidxB[s]);
        int r = s1 * AN_ + a1 * N_ + n1;
        int c = s2 * AN_ + a2 * N_ + n2;
        ooff[s] = (((size_t)k * F_ + r) * F_ + c) * 2;
    }
    for (int e = 0; e < E_; ++e) {
        int src = eidx[e], dst = eidx[E_ + e];
        bool dir = (src == a1) && (dst == a2);
        bool rev = (src == a2) && (dst == a1);
        if (!dir && !rev) continue;
        float ang = shift[e * 3 + 0] * kx + shift[e * 3 + 1] * ky + shift[e * 3 + 2] * kz;
        float sn, cs;
        sincosf(ang, &sn, &cs);
        float px = cs, py = -sn;
        const float* f = hop + (size_t)e * FEAT_;
        if (dir) {
#pragma unroll
            for (int s = 0; s < 3; ++s) {
                float hr = 0.5f * f[ridxA[s]], hi = 0.5f * f[iidxA[s]];
                accr[s] += hr * px - hi * py;
                acci[s] += hr * py + hi * px;
            }
        }
        if (rev) {
#pragma unroll
            for (int s = 0; s < 3; ++s) {
                float hr = 0.5f * f[ridxB[s]], hi = 0.5f * f[iidxB[s]];
                accr[s] += hr * px - hi * py;
                acci[s] -= hr * py + hi * px;
            }
        }
    }
    if (a1 == a2) {
        const float* f = ons + (size_t)a1 * FEAT_;
#pragma unroll
        for (int s = 0; s < 3; ++s) {
            accr[s] += 0.5f * (f[ridxA[s]] + f[ridxB[s]]);
            acci[s] += 0.5f * (f[iidxA[s]] - f[iidxB[s]]);
        }
    }
#pragma unroll
    for (int s = 0; s < 3; ++s) {
        if (valid[s]) {
            v2f v;
            v.x = accr[s];
            v.y = acci[s];
            __builtin_nontemporal_store(v, (v2f*)(out + ooff[s]));
        }
    }
}

extern "C" void kernel_launch(void* const* d_in, const int* in_sizes, int n_in,
                              void* d_out, int out_size, void* d_ws, size_t ws_size,
                              hipStream_t stream) {
    (void)in_sizes; (void)n_in; (void)out_size;
    const float* hop   = (const float*)d_in[0];  // [E, 648]
    const float* ons   = (const float*)d_in[1];  // [A, 648]
    const float* kpt   = (const float*)d_in[2];  // [K, 3]
    const int*   eidx  = (const int*)  d_in[3];  // [2, E]
    const float* shift = (const float*)d_in[4];  // [E, 3]
    float* out = (float*)d_out;                  // [K, F, F] complex64 as float pairs

    // scratch layout
    const size_t pw_bytes  = (size_t)E_ * K_ * sizeof(float2);   // 256 KB
    const size_t cnt_off   = pw_bytes;
    const size_t cnt_bytes = (size_t)PAIRS_ * sizeof(int);
    const size_t off_off   = cnt_off + cnt_bytes;
    const size_t off_bytes = (size_t)(PAIRS_ + 1) * sizeof(int);
    const size_t lst_off   = off_off + off_bytes;
    const size_t lst_bytes = (size_t)E_ * sizeof(int);
    const size_t need      = lst_off + lst_bytes;                // ~344 KB

    if (ws_size >= need) {
        float2* pw   = (float2*)((char*)d_ws);
        int* counts  = (int*)((char*)d_ws + cnt_off);
        int* offs    = (int*)((char*)d_ws + off_off);
        int* lists   = (int*)((char*)d_ws + lst_off);

        phase_kernel<<<E_ / 16, 32, 0, stream>>>(shift, kpt, pw);
        zero_counts_kernel<<<(PAIRS_ + 255) / 256, 256, 0, stream>>>(counts);
        count_kernel<<<(E_ + 255) / 256, 256, 0, stream>>>(eidx, counts);
        scan_kernel<<<1, 1024, 0, stream>>>(counts, offs);
        fill_kernel<<<E_ / 128, 256, 0, stream>>>(eidx, offs, lists);
        main_kernel<<<dim3(PAIRS_, K_), 128, 0, stream>>>(hop, ons, pw, offs, lists, out);
    } else {
        fallback_kernel<<<dim3(PAIRS_, K_), 128, 0, stream>>>(hop, ons, kpt, eidx, shift, out);
    }
}